// cg_interaction_90529320665785
// MI455X (gfx1250) — compile-verified
//
#include <hip/hip_runtime.h>
#include <math.h>

#define EDGES 65536
#define LOW_STRIDE 1152                 // 9 rows * 128 channels
#define ALL_BASE ((size_t)EDGES * LOW_STRIDE)

typedef __attribute__((ext_vector_type(2))) float v2f;
typedef __attribute__((ext_vector_type(8))) float v8f;

__device__ __forceinline__ v8f wmma4(v2f a, v2f b, v8f c) {
  // D = A(16x4 f32) * B(4x16 f32) + C(16x16 f32)
  return __builtin_amdgcn_wmma_f32_16x16x4_f32(false, a, false, b, (short)0, c, false, false);
}

// ====================== W3J table init (mirrors reference) ======================
__device__ __forceinline__ int imaxi(int a, int b) { return a > b ? a : b; }
__device__ __forceinline__ int imini(int a, int b) { return a < b ? a : b; }

__device__ double dfact(int n) { double r = 1.0; for (int i = 2; i <= n; ++i) r *= (double)i; return r; }

__device__ double su2_cg(int j1, int m1, int j2, int m2, int j3, int m3) {
  if (m3 != m1 + m2) return 0.0;
  double pref = sqrt((2.0 * j3 + 1.0) * dfact(j3 + j1 - j2) * dfact(j3 - j1 + j2) *
                     dfact(j1 + j2 - j3) / dfact(j1 + j2 + j3 + 1));
  pref *= sqrt(dfact(j3 + m3) * dfact(j3 - m3) * dfact(j1 - m1) * dfact(j1 + m1) *
               dfact(j2 - m2) * dfact(j2 + m2));
  int kmin = imaxi(0, imaxi(j2 - j3 - m1, j1 - j3 + m2));
  int kmax = imini(j1 + j2 - j3, imini(j1 - m1, j2 + m2));
  double s = 0.0;
  for (int k = kmin; k <= kmax; ++k) {
    double d = dfact(k) * dfact(j1 + j2 - j3 - k) * dfact(j1 - m1 - k) *
               dfact(j2 + m2 - k) * dfact(j3 - j2 + m1 + k) * dfact(j3 - j1 - m2 + k);
    s += ((k & 1) ? -1.0 : 1.0) / d;
  }
  return pref * s;
}

__device__ void fill_q(int l, double qr[7][7], double qi[7][7]) {
  for (int i = 0; i < 7; ++i)
    for (int j = 0; j < 7; ++j) { qr[i][j] = 0.0; qi[i][j] = 0.0; }
  const double s = 0.70710678118654752440;
  for (int mm = 1; mm <= l; ++mm) {           // m = -mm
    qr[l - mm][l + mm] = s;                   // 1/sqrt2
    qi[l - mm][l - mm] = -s;                  // -i/sqrt2
  }
  qr[l][l] = 1.0;
  for (int m = 1; m <= l; ++m) {
    double sg = (m & 1) ? -1.0 : 1.0;
    qr[l + m][l + m] = sg * s;                // (-1)^m/sqrt2
    qi[l + m][l - m] = sg * s;                // i(-1)^m/sqrt2
  }
  int ph = l & 3;                             // multiply all by (-i)^l
  if (ph) {
    for (int i = 0; i < 7; ++i)
      for (int j = 0; j < 7; ++j) {
        double r = qr[i][j], im = qi[i][j];
        if (ph == 1)      { qr[i][j] = im;  qi[i][j] = -r;  }
        else if (ph == 2) { qr[i][j] = -r;  qi[i][j] = -im; }
        else              { qr[i][j] = -im; qi[i][j] = r;   }
      }
  }
}

__global__ void w3j_init_kernel(float* __restrict__ ws) {
  const int t = threadIdx.x;
  if (t >= 23) return;
  const int TLA[23] = {0,1,2,3, 0,1,1,2,2,3, 0,1,1,2,2,3,3, 0,1,2,2,3,3};
  const int TLB[23] = {0,1,2,3, 1,0,2,1,3,2, 2,1,3,0,2,1,3, 3,2,1,3,0,2};
  const int TLO[23] = {0,0,0,0, 1,1,1,1,1,1, 2,2,2,2,2,2,2, 3,3,3,3,3,3};
  const int TOF[23] = {0,1,10,35, 84,93,102,147,192,297,
                       402,427,472,577,602,727,832, 1077,1126,1231,1336,1581,1630};
  const int la = TLA[t], lb = TLB[t], lo = TLO[t], off = TOF[t];
  const int d1 = 2*la+1, d2 = 2*lb+1, d3 = 2*lo+1;

  double C[7][7][7];
  for (int i = 0; i < 7; ++i)
    for (int j = 0; j < 7; ++j)
      for (int k = 0; k < 7; ++k) C[i][j][k] = 0.0;
  for (int m1 = -la; m1 <= la; ++m1)
    for (int m2 = -lb; m2 <= lb; ++m2) {
      int m3 = m1 + m2;
      if (m3 >= -lo && m3 <= lo)
        C[la + m1][lb + m2][lo + m3] = su2_cg(la, m1, lb, m2, lo, m3);
    }
  double q1r[7][7], q1i[7][7], q2r[7][7], q2i[7][7], q3r[7][7], q3i[7][7];
  fill_q(la, q1r, q1i); fill_q(lb, q2r, q2i); fill_q(lo, q3r, q3i);

  double norm2 = 0.0;
  for (int j = 0; j < d1; ++j)
    for (int l2 = 0; l2 < d2; ++l2)
      for (int nn = 0; nn < d3; ++nn) {
        double acc = 0.0;
        for (int i = 0; i < d1; ++i) {
          const double ar = q1r[i][j], ai = q1i[i][j];
          if (ar == 0.0 && ai == 0.0) continue;
          for (int k = 0; k < d2; ++k) {
            const double br = q2r[k][l2], bi = q2i[k][l2];
            if (br == 0.0 && bi == 0.0) continue;
            const double abr = ar * br - ai * bi;
            const double abi = ar * bi + ai * br;
            for (int m = 0; m < d3; ++m) {
              const double c = C[i][k][m];
              if (c != 0.0)  // Re(Q1*Q2*conj(Q3))*C
                acc += c * (abr * q3r[nn][m] + abi * q3i[nn][m]);
            }
          }
        }
        ws[off + (j * d2 + l2) * d3 + nn] = (float)acc;
        norm2 += acc * acc;
      }
  const double inv = 1.0 / sqrt(norm2);
  for (int idx = 0; idx < d1 * d2 * d3; ++idx)
    ws[off + idx] = (float)((double)ws[off + idx] * inv);
}

// ====================== main fused tensor-product kernel ======================
template <int LA, int LB, int LO>
__device__ __forceinline__ void compute_z(const float* __restrict__ xx, const float* __restrict__ yy,
                                          const float* __restrict__ w3, float* __restrict__ z) {
  constexpr int DA = 2 * LA + 1, DB = 2 * LB + 1, DK = 2 * LO + 1;
#pragma unroll
  for (int k = 0; k < DK; ++k) z[k] = 0.0f;
#pragma unroll
  for (int i = 0; i < DA; ++i) {
    const float xi = xx[LA * LA + i];
#pragma unroll
    for (int j = 0; j < DB; ++j) {
      const float p = xi * yy[LB * LB + j];
#pragma unroll
      for (int k = 0; k < DK; ++k)
        z[k] = fmaf(p, w3[(i * DB + j) * DK + k], z[k]);
    }
  }
}

// fast sigmoid / silu: v_exp_f32 + v_rcp_f32, avoids IEEE div_fixup/div_fmas chains
__device__ __forceinline__ float sigm_(float v) {
  return __builtin_amdgcn_rcpf(1.0f + __expf(-v));
}
__device__ __forceinline__ float silu_(float v) { return v * sigm_(v); }

__global__ void __launch_bounds__(256) cg_main_kernel(
    const float* __restrict__ x, const float* __restrict__ y,
    const float* __restrict__ w_cg, const float* __restrict__ b_cg,
    const float* __restrict__ w_all, const float* __restrict__ b_all,
    const float* __restrict__ w3j, float* __restrict__ out) {
  __shared__ float sW3J[1875];
  __shared__ float sWcg[3200];
  __shared__ float sBcg[384];
  __shared__ float sWall[35];
  __shared__ float sBall[4];
  const int tid = threadIdx.x;
  for (int i = tid; i < 1875; i += 256) sW3J[i] = w3j[i];
  for (int i = tid; i < 3200; i += 256) sWcg[i] = w_cg[i];
  for (int i = tid; i < 384;  i += 256) sBcg[i] = b_cg[i];
  if (tid < 35) sWall[tid] = w_all[tid];
  if (tid < 4)  sBall[tid] = b_all[tid];
  __syncthreads();

  const int lane = tid & 31;
  const int wv   = tid >> 5;
  const bool hi  = (lane & 16) != 0;
  const int n    = lane & 15;
  const int e0   = (blockIdx.x * 8 + wv) * 16;   // 16-edge tile per wave
  const int e    = e0 + n;                       // lane's edge (both halves)

  float xx[16], yy[16];
  {
    const float4* xv = (const float4*)(x + (size_t)e * 16);
    const float4* yv = (const float4*)(y + (size_t)e * 16);
#pragma unroll
    for (int i = 0; i < 4; ++i) {
      float4 a = xv[i]; xx[4*i] = a.x; xx[4*i+1] = a.y; xx[4*i+2] = a.z; xx[4*i+3] = a.w;
      float4 b = yv[i]; yy[4*i] = b.x; yy[4*i+1] = b.y; yy[4*i+2] = b.z; yy[4*i+3] = b.w;
    }
  }

  // ---- z features per lane (edge e) ----
  float z0[4];
  compute_z<0,0,0>(xx, yy, sW3J + 0,  &z0[0]);
  compute_z<1,1,0>(xx, yy, sW3J + 1,  &z0[1]);
  compute_z<2,2,0>(xx, yy, sW3J + 10, &z0[2]);
  compute_z<3,3,0>(xx, yy, sW3J + 35, &z0[3]);
  float z1[6][3];
  compute_z<0,1,1>(xx, yy, sW3J + 84,  z1[0]);
  compute_z<1,0,1>(xx, yy, sW3J + 93,  z1[1]);
  compute_z<1,2,1>(xx, yy, sW3J + 102, z1[2]);
  compute_z<2,1,1>(xx, yy, sW3J + 147, z1[3]);
  compute_z<2,3,1>(xx, yy, sW3J + 192, z1[4]);
  compute_z<3,2,1>(xx, yy, sW3J + 297, z1[5]);
  float z2[7][5];
  compute_z<0,2,2>(xx, yy, sW3J + 402, z2[0]);
  compute_z<1,1,2>(xx, yy, sW3J + 427, z2[1]);
  compute_z<1,3,2>(xx, yy, sW3J + 472, z2[2]);
  compute_z<2,0,2>(xx, yy, sW3J + 577, z2[3]);
  compute_z<2,2,2>(xx, yy, sW3J + 602, z2[4]);
  compute_z<3,1,2>(xx, yy, sW3J + 727, z2[5]);
  compute_z<3,3,2>(xx, yy, sW3J + 832, z2[6]);

  // ---- cg_all (mul = 1 entries, incl. lo = 3), lanes 0..15 only ----
  if (!hi) {
    float z3[6][7];
    compute_z<0,3,3>(xx, yy, sW3J + 1077, z3[0]);
    compute_z<1,2,3>(xx, yy, sW3J + 1126, z3[1]);
    compute_z<2,1,3>(xx, yy, sW3J + 1231, z3[2]);
    compute_z<2,3,3>(xx, yy, sW3J + 1336, z3[3]);
    compute_z<3,0,3>(xx, yy, sW3J + 1581, z3[4]);
    compute_z<3,2,3>(xx, yy, sW3J + 1630, z3[5]);

    float* oa = out + ALL_BASE + (size_t)e * 16;
    float s0 = 0.f;
#pragma unroll
    for (int p = 0; p < 4; ++p) s0 = fmaf(z0[p], sWall[p], s0);
    __builtin_nontemporal_store(silu_(0.5f * s0 + sBall[0]), oa + 0);
    float ga[3];
#pragma unroll
    for (int j = 0; j < 3; ++j) {
      float t = 0.f;
#pragma unroll
      for (int p = 0; p < 4; ++p) t = fmaf(z0[p], sWall[4 + 3 * p + j], t);
      ga[j] = sigm_(0.5f * t + sBall[1 + j]);
    }
#pragma unroll
    for (int k = 0; k < 3; ++k) {
      float t = 0.f;
#pragma unroll
      for (int p = 0; p < 6; ++p) t = fmaf(z1[p][k], sWall[16 + p], t);
      __builtin_nontemporal_store(0.70710678f * t * ga[0], oa + 1 + k);      // sqrt(3/6)
    }
#pragma unroll
    for (int k = 0; k < 5; ++k) {
      float t = 0.f;
#pragma unroll
      for (int p = 0; p < 7; ++p) t = fmaf(z2[p][k], sWall[22 + p], t);
      __builtin_nontemporal_store(0.84515425f * t * ga[1], oa + 4 + k);      // sqrt(5/7)
    }
#pragma unroll
    for (int k = 0; k < 7; ++k) {
      float t = 0.f;
#pragma unroll
      for (int p = 0; p < 6; ++p) t = fmaf(z3[p][k], sWall[29 + p], t);
      __builtin_nontemporal_store(1.0801234f * t * ga[2], oa + 9 + k);       // sqrt(7/6)
    }
  }

  // ---- cg_low via V_WMMA_F32_16X16X4_F32 ----
  const v8f c0 = {0.f, 0.f, 0.f, 0.f, 0.f, 0.f, 0.f, 0.f};
  v2f a0;  // A frag for lo=0 entries: K = 4 paths (la=lb diag dots)
  a0.x = hi ? z0[2] : z0[0];
  a0.y = hi ? z0[3] : z0[1];

  for (int ct = 0; ct < 8; ++ct) {
    const int ch = ct * 16 + n;   // channel 0..127

    // sc (ei0, base 0, stride 128) and gates (ei1, base 512, stride 256)
    v2f b;
    b.x = hi ? sWcg[256 + ch] : sWcg[ch];
    b.y = hi ? sWcg[384 + ch] : sWcg[128 + ch];
    v8f fsc = wmma4(a0, b, c0);
    b.x = hi ? sWcg[1024 + ch] : sWcg[512 + ch];
    b.y = hi ? sWcg[1280 + ch] : sWcg[768 + ch];
    v8f fg1 = wmma4(a0, b, c0);
    b.x = hi ? sWcg[1152 + ch] : sWcg[640 + ch];
    b.y = hi ? sWcg[1408 + ch] : sWcg[896 + ch];
    v8f fg2 = wmma4(a0, b, c0);

    const float bs  = sBcg[ch];
    const float bg1 = sBcg[128 + ch];
    const float bg2 = sBcg[256 + ch];
    float g1[8], g2[8];
#pragma unroll
    for (int r = 0; r < 8; ++r) {
      const int m = r + (hi ? 8 : 0);
      float* orow = out + (size_t)(e0 + m) * LOW_STRIDE;
      __builtin_nontemporal_store(silu_(0.5f * fsc[r] + bs), orow + ch);
      g1[r] = sigm_(0.5f * fg1[r] + bg1);
      g2[r] = sigm_(0.5f * fg2[r] + bg2);
    }

    // lo = 1 rows (3 comps, 6 paths, weight base 1536, stride 128)
    v2f b1a, b1b;
    b1a.x = hi ? sWcg[1536 + 256 + ch] : sWcg[1536 + ch];
    b1a.y = hi ? sWcg[1536 + 384 + ch] : sWcg[1536 + 128 + ch];
    b1b.x = hi ? 0.f : sWcg[1536 + 512 + ch];
    b1b.y = hi ? 0.f : sWcg[1536 + 640 + ch];
#pragma unroll
    for (int k = 0; k < 3; ++k) {
      v2f a;  a.x  = hi ? z1[2][k] : z1[0][k];  a.y  = hi ? z1[3][k] : z1[1][k];
      v8f acc = wmma4(a, b1a, c0);
      v2f a2; a2.x = hi ? 0.f : z1[4][k];       a2.y = hi ? 0.f : z1[5][k];
      acc = wmma4(a2, b1b, acc);
#pragma unroll
      for (int r = 0; r < 8; ++r) {
        const int m = r + (hi ? 8 : 0);
        float* orow = out + (size_t)(e0 + m) * LOW_STRIDE;
        __builtin_nontemporal_store(0.70710678f * acc[r] * g1[r], orow + (1 + k) * 128 + ch);
      }
    }

    // lo = 2 rows (5 comps, 7 paths, weight base 2304, stride 128)
    v2f b2a, b2b;
    b2a.x = hi ? sWcg[2304 + 256 + ch] : sWcg[2304 + ch];
    b2a.y = hi ? sWcg[2304 + 384 + ch] : sWcg[2304 + 128 + ch];
    b2b.x = hi ? sWcg[2304 + 768 + ch] : sWcg[2304 + 512 + ch];
    b2b.y = hi ? 0.f : sWcg[2304 + 640 + ch];
#pragma unroll
    for (int k = 0; k < 5; ++k) {
      v2f a;  a.x  = hi ? z2[2][k] : z2[0][k];  a.y  = hi ? z2[3][k] : z2[1][k];
      v8f acc = wmma4(a, b2a, c0);
      v2f a2; a2.x = hi ? z2[6][k] : z2[4][k];  a2.y = hi ? 0.f : z2[5][k];
      acc = wmma4(a2, b2b, acc);
#pragma unroll
      for (int r = 0; r < 8; ++r) {
        const int m = r + (hi ? 8 : 0);
        float* orow = out + (size_t)(e0 + m) * LOW_STRIDE;
        __builtin_nontemporal_store(0.84515425f * acc[r] * g2[r], orow + (4 + k) * 128 + ch);
      }
    }
  }
}

extern "C" void kernel_launch(void* const* d_in, const int* in_sizes, int n_in,
                              void* d_out, int out_size, void* d_ws, size_t ws_size,
                              hipStream_t stream) {
  const float* x     = (const float*)d_in[0];
  const float* y     = (const float*)d_in[1];
  const float* w_cg  = (const float*)d_in[2];
  const float* b_cg  = (const float*)d_in[3];
  const float* w_all = (const float*)d_in[4];
  const float* b_all = (const float*)d_in[5];
  float* out = (float*)d_out;
  float* w3j = (float*)d_ws;   // 1875 floats

  w3j_init_kernel<<<1, 32, 0, stream>>>(w3j);
  cg_main_kernel<<<EDGES / 128, 256, 0, stream>>>(x, y, w_cg, b_cg, w_all, b_all, w3j, out);
}